// S5SSM_8117488189490
// MI455X (gfx1250) — compile-verified
//
#include <hip/hip_runtime.h>

// S5 SSM forward for MI455X (gfx1250): fp32 WMMA GEMMs + LDS-resident chunked scan.
#define L_SEQ 32768
#define H_DIM 128
#define P_DIM 256
#define TC    64                 // timesteps per chunk
#define NCH   (L_SEQ / TC)       // 512 chunks
#define GRP   8                  // chunks per combine group
#define NG    (NCH / GRP)        // 64 groups

typedef __attribute__((ext_vector_type(2))) float v2f;
typedef __attribute__((ext_vector_type(8))) float v8f;

#if __has_builtin(__builtin_amdgcn_global_load_async_to_lds_b128)
#define S5_ASYNC_LDS 1
typedef int s5_v4i __attribute__((vector_size(16)));
typedef s5_v4i __attribute__((address_space(1)))* s5_gptr;   // global side
typedef s5_v4i __attribute__((address_space(3)))* s5_lptr;   // LDS side
#endif

__device__ __forceinline__ v8f wmma_f32(v2f a, v2f b, v8f c) {
  // V_WMMA_F32_16X16X4_F32 : D = A(16x4) * B(4x16) + C(16x16), all f32
  return __builtin_amdgcn_wmma_f32_16x16x4_f32(false, a, false, b, (short)0, c,
                                               false, false);
}

// (A,b) <- (c.A * A, c.A * b + c.b)   composition of affine recurrences
__device__ __forceinline__ void comb(float4& Ab, const float4 c) {
  const float Ar = c.x * Ab.x - c.y * Ab.y;
  const float Ai = c.x * Ab.y + c.y * Ab.x;
  const float br = c.x * Ab.z - c.y * Ab.w + c.z;
  const float bi = c.x * Ab.w + c.y * Ab.z + c.w;
  Ab = make_float4(Ar, Ai, br, bi);
}
__device__ __forceinline__ float2 apply_c(const float4 c, const float2 x) {
  return make_float2(c.x * x.x - c.y * x.y + c.z,
                     c.x * x.y + c.y * x.x + c.w);
}

// ---- transpose/pack B and C once: coalesced WMMA B-operand fragments ----
// Bt[h][p] = (Bre[p][h],  Bim[p][h])     (H x P, float2)
// Ct[p][h] = (Cre[h][p], -Cim[h][p])     (P x H, float2; minus folded in)
__global__ __launch_bounds__(256)
void s5_pack_kernel(const float* __restrict__ Bre, const float* __restrict__ Bim,
                    const float* __restrict__ Cre, const float* __restrict__ Cim,
                    float2* __restrict__ Bt, float2* __restrict__ Ct)
{
  const int idx = blockIdx.x * 256 + threadIdx.x;   // 0 .. H*P-1
  {
    const int h = idx / P_DIM, p = idx % P_DIM;
    Bt[idx] = make_float2(Bre[p * H_DIM + h], Bim[p * H_DIM + h]);
  }
  {
    const int p = idx / H_DIM, h = idx % H_DIM;
    Ct[idx] = make_float2(Cre[h * P_DIM + p], -Cim[h * P_DIM + p]);
  }
}

// One chunk of TC timesteps. phase==0: local scan from zero state, emit carries.
// phase==2: scan seeded with x_in, then WMMA readout + D*u epilogue.
__global__ __launch_bounds__(256)
void s5_chunk_kernel(const float* __restrict__ u,
                     const float* __restrict__ dt,
                     const float* __restrict__ Lre,
                     const float* __restrict__ Lim,
                     const float* __restrict__ lstep,
                     const float2* __restrict__ Bt,
                     const float2* __restrict__ Ct,
                     const float* __restrict__ Dv,
                     float* __restrict__ out,
                     float4* __restrict__ carries,
                     const float2* __restrict__ xin,
                     int phase)
{
  extern __shared__ char smem[];
  float2* xbuf = (float2*)smem;                                        // TC x P (Bu, then states)
  float*  ubuf = (float*)(smem + (size_t)TC * P_DIM * sizeof(float2)); // TC x H
  float*  dtb  = (float*)(smem + (size_t)TC * P_DIM * sizeof(float2)
                               + (size_t)TC * H_DIM * sizeof(float));  // TC

  const int tid   = threadIdx.x;
  const int wave  = tid >> 5;
  const int lane  = tid & 31;
  const int half  = lane >> 4;   // WMMA half-wave
  const int l16   = lane & 15;
  const int chunk = blockIdx.x;
  const int tbase = chunk * TC;

  // ---- stage u tile and dt into LDS ----
  {
    const float* usrc = u + (size_t)tbase * H_DIM;
#if S5_ASYNC_LDS
#pragma unroll
    for (int i = 0; i < (TC * H_DIM) / (256 * 4); ++i)   // 16B per lane per iter
      __builtin_amdgcn_global_load_async_to_lds_b128(
          (s5_gptr)(usrc + tid * 4 + i * 1024),
          (s5_lptr)(ubuf + tid * 4 + i * 1024), 0, 0);
#else
#pragma unroll
    for (int i = 0; i < (TC * H_DIM) / 256; ++i)
      ubuf[tid + i * 256] = usrc[tid + i * 256];
#endif
    if (tid < TC) dtb[tid] = dt[tbase + tid];
#if S5_ASYNC_LDS
#if __has_builtin(__builtin_amdgcn_s_wait_asynccnt)
    __builtin_amdgcn_s_wait_asynccnt(0);
#else
    asm volatile("s_wait_asynccnt 0" ::: "memory");
#endif
#endif
  }
  __syncthreads();

  // ---- GEMM1: Bu_raw[t][p] = sum_h u[t][h] * B[p][h]  (re & im together) ----
  // Wave owns a 16-wide p-strip and sweeps all 4 row tiles: each coalesced
  // Bt fragment pair feeds 8 v_wmma ops.
  for (int nt = wave; nt < P_DIM / 16; nt += 8) {
    const int n0 = nt * 16;
    v8f accr[TC / 16] = {};
    v8f acci[TC / 16] = {};
    for (int k = 0; k < H_DIM; k += 4) {
      const int kk = k + 2 * half;
      const float2 b0 = Bt[(size_t)kk * P_DIM + n0 + l16];        // row kk (re,im)
      const float2 b1 = Bt[(size_t)(kk + 1) * P_DIM + n0 + l16];  // row kk+1
      v2f br, bi;
      br.x = b0.x; br.y = b1.x;
      bi.x = b0.y; bi.y = b1.y;
#pragma unroll
      for (int mt = 0; mt < TC / 16; ++mt) {
        const float2 av = *(const float2*)&ubuf[(mt * 16 + l16) * H_DIM + kk];
        v2f a; a.x = av.x; a.y = av.y;
        accr[mt] = wmma_f32(a, br, accr[mt]);
        acci[mt] = wmma_f32(a, bi, acci[mt]);
      }
    }
#pragma unroll
    for (int mt = 0; mt < TC / 16; ++mt)
#pragma unroll
      for (int v = 0; v < 8; ++v) {
        const int row = mt * 16 + v + 8 * half;
        xbuf[row * P_DIM + n0 + l16] = make_float2(accr[mt][v], acci[mt][v]);
      }
  }
  __syncthreads();

  // ---- local recurrence: thread p owns complex state p; in-place Bu -> x ----
  {
    const int p = tid;
    const float lr = Lre[p], li = Lim[p];
    const float st = __expf(lstep[p]);
    const float a2 = lr * lr + li * li;
    const bool useg = (a2 > 1e-12f);          // |Lambda| > eps
    const float i2 = useg ? (1.0f / a2) : 0.0f;
    float xr, xi;
    if (phase == 0) { xr = 0.f; xi = 0.f; }
    else { const float2 s = xin[(size_t)chunk * P_DIM + p]; xr = s.x; xi = s.y; }
    float Ar = 1.f, Ai = 0.f;
    for (int t = 0; t < TC; ++t) {
      const float dlt = dtb[t] * st;
      const float e = __expf(lr * dlt);
      float sn, cs;
      __sincosf(li * dlt, &sn, &cs);
      const float lbr = e * cs, lbi = e * sn;       // Lambda_bar
      float gr, gi;                                 // gamma = (Lb-1)/Lambda
      if (useg) {
        const float nr = lbr - 1.0f, ni = lbi;
        gr = (nr * lr + ni * li) * i2;
        gi = (ni * lr - nr * li) * i2;
      } else { gr = dlt; gi = 0.0f; }
      const float2 bu = xbuf[t * P_DIM + p];
      const float br = gr * bu.x - gi * bu.y;       // gamma * (B@u)
      const float bi = gr * bu.y + gi * bu.x;
      const float nxr = lbr * xr - lbi * xi + br;   // x = Lb*x + Bu
      const float nxi = lbr * xi + lbi * xr + bi;
      xr = nxr; xi = nxi;
      xbuf[t * P_DIM + p] = make_float2(xr, xi);
      if (phase == 0) {                             // Atot = Lb * Atot
        const float nAr = lbr * Ar - lbi * Ai;
        Ai = lbr * Ai + lbi * Ar;
        Ar = nAr;
      }
    }
    if (phase == 0)
      carries[(size_t)chunk * P_DIM + p] = make_float4(Ar, Ai, xr, xi);
  }

  if (phase == 0) return;
  __syncthreads();

  // ---- GEMM2: y[t][h] = 2*(xr@Cre^T - xi@Cim^T) + u[t][h]*D[h] ----
  // 8 waves cover H=128: each wave owns one 16-wide h-strip, all 4 row tiles.
  {
    const int n0 = wave * 16;
    v8f acc[TC / 16] = {};
    for (int k = 0; k < P_DIM; k += 4) {
      const int kk = k + 2 * half;
      const float2 c0 = Ct[(size_t)kk * H_DIM + n0 + l16];        // (Cre, -Cim)
      const float2 c1 = Ct[(size_t)(kk + 1) * H_DIM + n0 + l16];
      v2f br, bi;
      br.x = c0.x; br.y = c1.x;
      bi.x = c0.y; bi.y = c1.y;
#pragma unroll
      for (int mt = 0; mt < TC / 16; ++mt) {
        const float4 xa = *(const float4*)&xbuf[(mt * 16 + l16) * P_DIM + kk];
        v2f ar, ai;
        ar.x = xa.x; ar.y = xa.z;     // real parts of states kk,kk+1
        ai.x = xa.y; ai.y = xa.w;     // imag parts
        acc[mt] = wmma_f32(ar, br, acc[mt]);
        acc[mt] = wmma_f32(ai, bi, acc[mt]);
      }
    }
    const int col = n0 + l16;
    const float dcol = Dv[col];
#pragma unroll
    for (int mt = 0; mt < TC / 16; ++mt)
#pragma unroll
      for (int v = 0; v < 8; ++v) {
        const int row = mt * 16 + v + 8 * half;
        out[(size_t)(tbase + row) * H_DIM + col] =
            2.0f * acc[mt][v] + ubuf[row * H_DIM + col] * dcol;
      }
  }
}

// ---- 3-level cross-chunk combine (sequential depth 8+64+8, batched loads) ----
__global__ __launch_bounds__(256)
void s5_group_reduce(const float4* __restrict__ carries, float4* __restrict__ gcar)
{
  const int p = threadIdx.x, g = blockIdx.x;
  float4 cb[GRP];
#pragma unroll
  for (int i = 0; i < GRP; ++i) cb[i] = carries[(size_t)(g * GRP + i) * P_DIM + p];
  float4 acc = make_float4(1.f, 0.f, 0.f, 0.f);
#pragma unroll
  for (int i = 0; i < GRP; ++i) comb(acc, cb[i]);
  gcar[(size_t)g * P_DIM + p] = acc;
}

__global__ __launch_bounds__(256)
void s5_group_scan(const float4* __restrict__ gcar, float2* __restrict__ gx, int ng)
{
  const int p = threadIdx.x;
  float2 x = make_float2(0.f, 0.f);
  for (int g0 = 0; g0 < ng; g0 += GRP) {
    float4 cb[GRP];
#pragma unroll
    for (int i = 0; i < GRP; ++i) cb[i] = gcar[(size_t)(g0 + i) * P_DIM + p];
#pragma unroll
    for (int i = 0; i < GRP; ++i) {
      gx[(size_t)(g0 + i) * P_DIM + p] = x;
      x = apply_c(cb[i], x);
    }
  }
}

__global__ __launch_bounds__(256)
void s5_group_expand(const float4* __restrict__ carries,
                     const float2* __restrict__ gx, float2* __restrict__ xin)
{
  const int p = threadIdx.x, g = blockIdx.x;
  float4 cb[GRP];
#pragma unroll
  for (int i = 0; i < GRP; ++i) cb[i] = carries[(size_t)(g * GRP + i) * P_DIM + p];
  float2 x = gx[(size_t)g * P_DIM + p];
#pragma unroll
  for (int i = 0; i < GRP; ++i) {
    xin[(size_t)(g * GRP + i) * P_DIM + p] = x;
    x = apply_c(cb[i], x);
  }
}

extern "C" void kernel_launch(void* const* d_in, const int* in_sizes, int n_in,
                              void* d_out, int out_size, void* d_ws, size_t ws_size,
                              hipStream_t stream) {
  const float* u    = (const float*)d_in[0];
  const float* dt   = (const float*)d_in[1];
  const float* Lre  = (const float*)d_in[2];
  const float* Lim  = (const float*)d_in[3];
  const float* lst  = (const float*)d_in[4];
  const float* Bre  = (const float*)d_in[5];
  const float* Bim  = (const float*)d_in[6];
  const float* Cre  = (const float*)d_in[7];
  const float* Cim  = (const float*)d_in[8];
  const float* Dv   = (const float*)d_in[9];
  float* out = (float*)d_out;

  char* ws = (char*)d_ws;
  float4* carries = (float4*)ws;                         ws += (size_t)NCH * P_DIM * sizeof(float4); // 2 MB
  float2* xin     = (float2*)ws;                         ws += (size_t)NCH * P_DIM * sizeof(float2); // 1 MB
  float4* gcar    = (float4*)ws;                         ws += (size_t)NG  * P_DIM * sizeof(float4); // 256 KB
  float2* gx      = (float2*)ws;                         ws += (size_t)NG  * P_DIM * sizeof(float2); // 128 KB
  float2* Bt      = (float2*)ws;                         ws += (size_t)H_DIM * P_DIM * sizeof(float2); // 256 KB
  float2* Ct      = (float2*)ws;                         // 256 KB

  const size_t shmem = (size_t)TC * P_DIM * sizeof(float2)   // 128 KB states
                     + (size_t)TC * H_DIM * sizeof(float)    //  32 KB u tile
                     + (size_t)TC * sizeof(float);           //  dt

  s5_pack_kernel<<<dim3((H_DIM * P_DIM) / 256), dim3(256), 0, stream>>>(
      Bre, Bim, Cre, Cim, Bt, Ct);
  s5_chunk_kernel<<<dim3(NCH), dim3(256), shmem, stream>>>(
      u, dt, Lre, Lim, lst, Bt, Ct, Dv, out, carries, xin, 0);
  s5_group_reduce<<<dim3(NG), dim3(256), 0, stream>>>(carries, gcar);
  s5_group_scan<<<dim3(1), dim3(256), 0, stream>>>(gcar, gx, NG);
  s5_group_expand<<<dim3(NG), dim3(256), 0, stream>>>(carries, gx, xin);
  s5_chunk_kernel<<<dim3(NCH), dim3(256), shmem, stream>>>(
      u, dt, Lre, Lim, lst, Bt, Ct, Dv, out, carries, xin, 2);
}